// SphericalHarmonicsEmbedding_72645076844660
// MI455X (gfx1250) — compile-verified
//
#include <hip/hip_runtime.h>
#include <hip/hip_bf16.h>

typedef float v2f __attribute__((ext_vector_type(2)));
typedef float v8f __attribute__((ext_vector_type(8)));

#define SH_L      4
#define NMONO     35          // monomials of degree <= 4 in x,y,z
#define KPAD      36          // padded to 9 chunks of K=4
#define NCH       50          // 2*(L+1)^2 output channels
#define NPAD      64          // padded to 4 tiles of N=16
#define ITERS     8           // 16-point groups per wave

__device__ __forceinline__ int mono_idx(int a, int b, int c) {
    int d = a + b + c;
    int base = (d == 0) ? 0 : (d == 1) ? 1 : (d == 2) ? 4 : (d == 3) ? 10 : 20;
    int pos = 0;
    for (int aa = d; aa > a; --aa) pos += (d - aa + 1); // entries for higher a
    pos += (d - a) - b;                                  // b runs d-a .. 0
    return base + pos;
}

// ---------------------------------------------------------------------------
// Setup: build the 36x64 monomial->channel matrix C (row-major [k][n]) in f64,
// mirroring the reference's Legendre recurrences on z-polynomial coefficients.
// ---------------------------------------------------------------------------
__global__ void sh_build_coeff(float* __restrict__ C) {
    if (threadIdx.x != 0 || blockIdx.x != 0) return;
    for (int i = 0; i < KPAD * NPAD; ++i) C[i] = 0.0f;

    const double PI = 3.14159265358979323846;
    // pb[l][m][k] = coeff of z^k in Pbar_l^m(z) = P_l^m / st^m  (Condon-Shortley)
    double pb[SH_L + 1][SH_L + 1][SH_L + 1];
    for (int l = 0; l <= SH_L; ++l)
        for (int m = 0; m <= SH_L; ++m)
            for (int k = 0; k <= SH_L; ++k) pb[l][m][k] = 0.0;
    pb[0][0][0] = 1.0;
    for (int m = 1; m <= SH_L; ++m)                       // diagonal: Pbar_m^m = -(2m-1) Pbar_{m-1}^{m-1}
        for (int k = 0; k <= SH_L; ++k)
            pb[m][m][k] = -(2.0 * m - 1.0) * pb[m - 1][m - 1][k];
    for (int m = 0; m < SH_L; ++m)                        // subdiagonal: Pbar_{m+1}^m = (2m+1) z Pbar_m^m
        for (int k = SH_L; k >= 1; --k)
            pb[m + 1][m][k] = (2.0 * m + 1.0) * pb[m][m][k - 1];
    for (int m = 0; m <= SH_L; ++m)                       // (l-m)Pbar_l^m = (2l-1) z Pbar_{l-1}^m - (l+m-1) Pbar_{l-2}^m
        for (int l = m + 2; l <= SH_L; ++l)
            for (int k = 0; k <= SH_L; ++k) {
                double t = (2.0 * l - 1.0) * (k >= 1 ? pb[l - 1][m][k - 1] : 0.0)
                         - (l + m - 1.0) * pb[l - 2][m][k];
                pb[l][m][k] = t / (double)(l - m);
            }

    double fact[9];
    fact[0] = 1.0;
    for (int i = 1; i <= 8; ++i) fact[i] = fact[i - 1] * i;

    int ch = 0;
    for (int l = 0; l <= SH_L; ++l) {
        for (int mm = -l; mm <= l; ++mm) {
            int am = mm < 0 ? -mm : mm;
            double Nn = sqrt((2.0 * l + 1.0) / (4.0 * PI) * fact[l - am] / fact[l + am]);
            double sre = 1.0, sim = 1.0;
            if (mm < 0) { double s = (am & 1) ? -1.0 : 1.0; sre = s; sim = -s; }
            // channel_re = N * Pbar(z) * Re((x+iy)^am), channel_im = N * Pbar(z) * Im((x+iy)^am)
            for (int k = 0; k <= SH_L; ++k) {
                double pc = pb[l][am][k];
                if (pc == 0.0) continue;
                for (int j = 0; j <= am; ++j) {           // (x+iy)^am = sum_j C(am,j) i^j x^{am-j} y^j
                    double Cb = fact[am] / (fact[j] * fact[am - j]);
                    int idx = mono_idx(am - j, j, k);
                    if ((j & 1) == 0) {
                        double s = ((j / 2) & 1) ? -1.0 : 1.0;
                        C[idx * NPAD + ch]     += (float)(Nn * pc * Cb * s * sre);
                    } else {
                        double s = (((j - 1) / 2) & 1) ? -1.0 : 1.0;
                        C[idx * NPAD + ch + 1] += (float)(Nn * pc * Cb * s * sim);
                    }
                }
            }
            ch += 2;
        }
    }
}

// ---------------------------------------------------------------------------
// Main: per wave, 16 points/group; monomials on VALU, projection on WMMA f32.
// A fragment (16x4 f32): lanes 0-15 = rows, V0/V1 = K0/K1; lanes 16-31 = K2/K3.
// D tile (16x16 f32): lane L, VGPR v -> (row = v + 8*(L>=16), col = L&15).
// All monomials are NAMED scalars (no alloca -> no LDS promotion).
// ---------------------------------------------------------------------------
__global__ __launch_bounds__(256) void sh_embed_kernel(const float* __restrict__ dirs,
                                                       const float* __restrict__ C,
                                                       float* __restrict__ out,
                                                       int npoints) {
    const int tid  = blockIdx.x * blockDim.x + threadIdx.x;
    const int lane = threadIdx.x & 31;
    const int wid  = tid >> 5;
    const int prow = lane & 15;
    const bool hiHalf = lane >= 16;

    // B fragments, loaded once per wave (L2-resident constant matrix)
    float b0[9][4], b1[9][4];
#pragma unroll
    for (int c = 0; c < 9; ++c) {
        int r0 = 4 * c + (hiHalf ? 2 : 0);
#pragma unroll
        for (int t = 0; t < 4; ++t) {
            int col = t * 16 + prow;
            b0[c][t] = C[r0 * NPAD + col];
            b1[c][t] = C[(r0 + 1) * NPAD + col];
        }
    }

    for (int it = 0; it < ITERS; ++it) {
        int group = wid * ITERS + it;
        int base  = group * 16;
        int point = base + prow;
        int pcl   = point < npoints ? point : (npoints - 1); // keep EXEC full for WMMA
        const float* p = dirs + pcl * 3;
        float x = p[0], y = p[1], z = p[2];

        // Named-scalar monomials in the canonical enumeration order
        float x2 = x * x, x3 = x2 * x, x4 = x3 * x;
        float y2 = y * y, y3 = y2 * y, y4 = y3 * y;
        float z2 = z * z, z3 = z2 * z, z4 = z3 * z;
        float xy = x * y, xz = x * z, yz = y * z, xyz = xy * z;

        float m00 = 1.0f;
        float m01 = x,       m02 = y,       m03 = z;
        float m04 = x2,      m05 = xy,      m06 = xz;
        float m07 = y2,      m08 = yz,      m09 = z2;
        float m10 = x3,      m11 = x2 * y,  m12 = x2 * z;
        float m13 = x * y2,  m14 = xyz,     m15 = x * z2;
        float m16 = y3,      m17 = y2 * z,  m18 = y * z2, m19 = z3;
        float m20 = x4,      m21 = x3 * y,  m22 = x3 * z;
        float m23 = x2 * y2, m24 = x2 * yz, m25 = x2 * z2;
        float m26 = x * y3,  m27 = xyz * y, m28 = xyz * z, m29 = x * z3;
        float m30 = y4,      m31 = y3 * z,  m32 = y2 * z2, m33 = y * z3, m34 = z4;
        float m35 = 0.0f;

        v8f acc0 = {}, acc1 = {}, acc2 = {}, acc3 = {};

#define SH_CHUNK(E0, E1, E2, E3)                                               \
        {                                                                      \
            float a0 = hiHalf ? (E2) : (E0);                                   \
            float a1 = hiHalf ? (E3) : (E1);                                   \
            v2f Af = {a0, a1};                                                 \
            v2f Bf0 = {b0[_c][0], b1[_c][0]};                                  \
            v2f Bf1 = {b0[_c][1], b1[_c][1]};                                  \
            v2f Bf2 = {b0[_c][2], b1[_c][2]};                                  \
            v2f Bf3 = {b0[_c][3], b1[_c][3]};                                  \
            acc0 = __builtin_amdgcn_wmma_f32_16x16x4_f32(false, Af, false, Bf0, (short)0, acc0, false, false); \
            acc1 = __builtin_amdgcn_wmma_f32_16x16x4_f32(false, Af, false, Bf1, (short)0, acc1, false, false); \
            acc2 = __builtin_amdgcn_wmma_f32_16x16x4_f32(false, Af, false, Bf2, (short)0, acc2, false, false); \
            acc3 = __builtin_amdgcn_wmma_f32_16x16x4_f32(false, Af, false, Bf3, (short)0, acc3, false, false); \
        }

        { enum { _c = 0 }; SH_CHUNK(m00, m01, m02, m03) }
        { enum { _c = 1 }; SH_CHUNK(m04, m05, m06, m07) }
        { enum { _c = 2 }; SH_CHUNK(m08, m09, m10, m11) }
        { enum { _c = 3 }; SH_CHUNK(m12, m13, m14, m15) }
        { enum { _c = 4 }; SH_CHUNK(m16, m17, m18, m19) }
        { enum { _c = 5 }; SH_CHUNK(m20, m21, m22, m23) }
        { enum { _c = 6 }; SH_CHUNK(m24, m25, m26, m27) }
        { enum { _c = 7 }; SH_CHUNK(m28, m29, m30, m31) }
        { enum { _c = 8 }; SH_CHUNK(m32, m33, m34, m35) }
#undef SH_CHUNK

        // Streaming NT stores: per instruction each half-wave covers 64 contiguous bytes
        const v8f accs[4] = {acc0, acc1, acc2, acc3};
#pragma unroll
        for (int t = 0; t < 4; ++t) {
            int chan = t * 16 + prow;
            if (chan < NCH) {
#pragma unroll
                for (int v = 0; v < 8; ++v) {
                    int row = base + v + (hiHalf ? 8 : 0);
                    if (row < npoints)
                        __builtin_nontemporal_store(accs[t][v], out + row * NCH + chan);
                }
            }
        }
    }
}

extern "C" void kernel_launch(void* const* d_in, const int* in_sizes, int n_in,
                              void* d_out, int out_size, void* d_ws, size_t ws_size,
                              hipStream_t stream) {
    const float* dirs = (const float*)d_in[0];
    float* out = (float*)d_out;
    float* C   = (float*)d_ws; // 36*64*4 = 9216 bytes

    int npoints = in_sizes[0] / 3;

    sh_build_coeff<<<dim3(1), dim3(1), 0, stream>>>(C);

    long groups  = (npoints + 15) / 16;
    long waves   = (groups + ITERS - 1) / ITERS;
    long threads = waves * 32;
    int  blocks  = (int)((threads + 255) / 255 > 0 ? (threads + 255) / 256 : 1);
    sh_embed_kernel<<<dim3(blocks), dim3(256), 0, stream>>>(dirs, C, out, npoints);
}